// PCF_5068061409533
// MI455X (gfx1250) — compile-verified
//
#include <hip/hip_runtime.h>
#include <math.h>

typedef float v2f __attribute__((ext_vector_type(2)));
typedef float v8f __attribute__((ext_vector_type(8)));

#define NBBINS 50
#define MAXNB  1024
#define PI_F      3.14159265358979f
#define TWO_PI_F  6.28318530717959f
#define INV_2PI_F 0.159154943091895f
#define GF_F      2.25675833419103f   /* 1/(sqrt(pi)*0.25) */
#define INV_S2    16.0f               /* 1/sigma^2, sigma = 0.25 */
#define DIST_CUT  7.7f                /* c_k<=5, exp(-16*2.7^2) underflows f32 */

__device__ __forceinline__ float shflx(float v, int mask) {
  return __shfl_xor(v, mask, 32);
}

__device__ __forceinline__ float fast_sqrtf(float x) {
  return __builtin_amdgcn_sqrtf(x);   // single v_sqrt_f32
}

__global__ void __launch_bounds__(128)
pcf_main_kernel(const float* __restrict__ da, const float* __restrict__ db,
                const int* __restrict__ scp, float* __restrict__ part,
                int na, int nb, float rmax) {
  __shared__ float2 sxy[MAXNB];   // {x, y}
  __shared__ float2 srq[MAXNB];   // {r, x*x + y*y}
  __shared__ float  sstats[4][NBBINS][3];

  const int tid = threadIdx.x;
  for (int j = tid; j < nb; j += blockDim.x) {
    float x = db[3 * j + 0], y = db[3 * j + 1], r = db[3 * j + 2];
    sxy[j] = make_float2(x, y);
    srq[j] = make_float2(r, x * x + y * y);
  }
  __syncthreads();

  const int wave   = tid >> 5;
  const int lane   = tid & 31;
  const int li     = lane & 15;
  const bool inLow = lane < 16;
  const int ibase  = (blockIdx.x * 4 + wave) * 16;
  const bool wave_valid = ibase < na;
  const int i      = min(ibase + li, na - 1);

  const float xa = da[3 * i + 0];
  const float ya = da[3 * i + 1];
  const float ra = da[3 * i + 2];
  const float qa = xa * xa + ya * ya;
  const float inv_rmax = 1.0f / rmax;

  // B matrix (4x16): lanes 0-15 hold rows K0/K1 = (xa, ya); lanes 16-31 rows K2/K3 = 0
  v2f Bm;
  Bm.x = inLow ? xa : 0.0f;
  Bm.y = inLow ? ya : 0.0f;

  float acc[NBBINS];
#pragma unroll
  for (int k = 0; k < NBBINS; ++k) acc[k] = 0.0f;

  if (wave_valid) {
    for (int jt = 0; jt < nb; jt += 16) {
      // Unconditional ds_load_b64, then lane-select (avoids exec-predicated loads)
      float2 xyj = sxy[jt + li];
      v2f Am;
      Am.x = inLow ? xyj.x : 0.0f;
      Am.y = inLow ? xyj.y : 0.0f;
      v8f D = {};
      // D[M=j, N=i] = xb_j*xa_i + yb_j*ya_i   (full f32 Gram tile)
      D = __builtin_amdgcn_wmma_f32_16x16x4_f32(false, Am, false, Bm,
                                                (short)0, D, false, false);
      const int jo = jt + (inLow ? 0 : 8);
#pragma unroll
      for (int r = 0; r < 8; ++r) {
        const int j = jo + r;
        float2 rq = srq[j];
        float g  = D[r];
        float d2 = fmaxf(qa + rq.y - 2.0f * g, 0.0f);
        float d  = fast_sqrtf(d2);
        float rb = rq.x;
        float r1 = fmaxf(ra, rb), r2 = fminf(ra, rb);
        float extent  = fmaxf(d + r1 + r2, 2.0f * r1);
        float overlap = fmaxf(r1 + r2 - d, 0.0f);
        float dist = (extent - overlap + d + r1 - r2) * inv_rmax;
        if (dist < DIST_CUT) {
#pragma unroll
          for (int k = 0; k < NBBINS; ++k) {
            float t = 0.1f * (float)(k + 1) - dist;
            acc[k] += __expf(-INV_S2 * t * t);
          }
        }
      }
    }
  }

  // Fuse the two half-waves: lanes L and L+16 share i, cover disjoint j halves
#pragma unroll
  for (int k = 0; k < NBBINS; ++k) acc[k] += shflx(acc[k], 16);

  // same_category: remove diagonal term j == i
  if (wave_valid && scp[0] != 0 && i < nb) {
    float2 xyi = sxy[i];
    float2 rqi = srq[i];
    float g  = xa * xyi.x + ya * xyi.y;
    float d2 = fmaxf(qa + rqi.y - 2.0f * g, 0.0f);
    float d  = fast_sqrtf(d2);
    float rb = rqi.x;
    float r1 = fmaxf(ra, rb), r2 = fminf(ra, rb);
    float extent  = fmaxf(d + r1 + r2, 2.0f * r1);
    float overlap = fmaxf(r1 + r2 - d, 0.0f);
    float dist = (extent - overlap + d + r1 - r2) * inv_rmax;
    if (dist < DIST_CUT) {
#pragma unroll
      for (int k = 0; k < NBBINS; ++k) {
        float t = 0.1f * (float)(k + 1) - dist;
        acc[k] -= __expf(-INV_S2 * t * t);
      }
    }
  }

  if (inLow) {
    // Perimeter weight per i: 4 domain edges; k-independent atan2 terms precomputed
    float sdx[4] = { xa, 1.0f - xa, ya, 1.0f - ya };
    float sdy[4] = { ya, ya, xa, xa };
    float t1[4], t2[4];
#pragma unroll
    for (int s = 0; s < 4; ++s) {
      t1[s] = atan2f(sdy[s], sdx[s]);
      t2[s] = atan2f(1.0f - sdy[s], sdx[s]);
    }
    const float inv_nb = 1.0f / (float)nb;
#pragma unroll
    for (int k = 0; k < NBBINS; ++k) {
      float rk = 0.1f * (float)(k + 1) * rmax;
      float full = TWO_PI_F;
#pragma unroll
      for (int s = 0; s < 4; ++s) {
        float ratio = fminf(fmaxf(sdx[s] / rk, -1.0f), 1.0f);
        float alpha = acosf(ratio);
        float sub = fminf(alpha, t1[s]) + fminf(alpha, t2[s]);
        full -= (rk > sdx[s]) ? sub : 0.0f;
      }
      float peri = fminf(fmaxf(full * INV_2PI_F, 0.0f), 1.0f);
      float w = (peri > 0.0f) ? (1.0f / peri) : 0.0f;
      float outer = rk + 0.5f * rmax;
      float inner = fmaxf(0.0f, rk - 0.5f * rmax);
      float area = PI_F * (outer * outer - inner * inner);
      float dens = acc[k] * GF_F * w / area * inv_nb;
      if (!wave_valid) dens = 0.0f;

      float sm = dens, mn = dens, mx = dens;
#pragma unroll
      for (int m = 8; m >= 1; m >>= 1) {
        sm += shflx(sm, m);
        mn = fminf(mn, shflx(mn, m));
        mx = fmaxf(mx, shflx(mx, m));
      }
      if (li == 0) {
        if (!wave_valid) { sm = 0.0f; mn = 3.4e38f; mx = -3.4e38f; }
        sstats[wave][k][0] = sm;
        sstats[wave][k][1] = mn;
        sstats[wave][k][2] = mx;
      }
    }
  }
  __syncthreads();

  if (tid < NBBINS) {
    float sm = 0.0f, mn = 3.4e38f, mx = -3.4e38f;
#pragma unroll
    for (int w = 0; w < 4; ++w) {
      sm += sstats[w][tid][0];
      mn = fminf(mn, sstats[w][tid][1]);
      mx = fmaxf(mx, sstats[w][tid][2]);
    }
    float* p = part + blockIdx.x * (3 * NBBINS);
    p[tid]              = sm;
    p[NBBINS + tid]     = mn;
    p[2 * NBBINS + tid] = mx;
  }
}

__global__ void __launch_bounds__(64)
pcf_final_kernel(const float* __restrict__ part, float* __restrict__ out,
                 int nblk, int na) {
  const int k = threadIdx.x;
  if (k >= NBBINS) return;
  float sm = 0.0f, mn = 3.4e38f, mx = -3.4e38f;
  for (int b = 0; b < nblk; ++b) {
    const float* p = part + b * (3 * NBBINS);
    sm += p[k];
    mn = fminf(mn, p[NBBINS + k]);
    mx = fmaxf(mx, p[2 * NBBINS + k]);
  }
  out[2 * k]     = 0.1f * (float)(k + 1);   // rs / RMAX
  out[2 * k + 1] = sm / (float)na;          // pcf mean
  out[2 * NBBINS + k]     = mn;             // pcf lower
  out[3 * NBBINS + k]     = mx;             // pcf upper
}

extern "C" void kernel_launch(void* const* d_in, const int* in_sizes, int n_in,
                              void* d_out, int out_size, void* d_ws, size_t ws_size,
                              hipStream_t stream) {
  (void)n_in; (void)out_size; (void)ws_size;
  const float* da = (const float*)d_in[0];
  const float* db = (const float*)d_in[1];
  const int*   sc = (const int*)d_in[2];
  float* out  = (float*)d_out;
  float* part = (float*)d_ws;

  const int na = in_sizes[0] / 3;
  const int nb = in_sizes[1] / 3;
  // RMAX depends only on the fixed NPOINTS=1024 of the reference
  const float rmax = 2.0f * sqrtf(1.0f / (2.0f * sqrtf(3.0f) * 1024.0f));

  const int ntiles = (na + 15) / 16;
  const int nblk   = (ntiles + 3) / 4;   // 4 waves (i-tiles) per block

  pcf_main_kernel<<<nblk, 128, 0, stream>>>(da, db, sc, part, na, nb, rmax);
  pcf_final_kernel<<<1, 64, 0, stream>>>(part, out, nblk, na);
}